// GraphSAGElayer_6665789243398
// MI455X (gfx1250) — compile-verified
//
#include <hip/hip_runtime.h>

// ---------------------------------------------------------------------------
// GraphSAGE layer for MI455X (gfx1250, wave32).
//   out[0 : 50000*128)        = relu([nh | scatter_mean(nh[src]->dst)] @ W1 + b1) @ W2 + b2
//   out[50000*128 : end)      = eh (passthrough)
// fp32 end-to-end; matrix work on V_WMMA_F32_16X16X4_F32 (fp32 matrix cores).
// ---------------------------------------------------------------------------

typedef float v2f __attribute__((ext_vector_type(2)));
typedef float v8f __attribute__((ext_vector_type(8)));

#define N_NODES 50000
#define N_EDGES 600000
#define D_IN    128
#define D_HID   256
#define D_OUT   128
#define XPITCH  260   // 256 + 4 pad: row stride ≡ 4 mod 64 banks -> 16 rows hit 16 distinct banks

// ---------------------- workspace zeroing ----------------------------------
__global__ void zero_ws_kernel(float* __restrict__ ws, int n) {
  int i = blockIdx.x * blockDim.x + threadIdx.x;
  if (i < n) ws[i] = 0.0f;
}

// ---------------------- eh passthrough (614 MB stream) ---------------------
__global__ void copy_eh_kernel(const float4* __restrict__ src,
                               float4* __restrict__ dst, int n4) {
  int i = blockIdx.x * blockDim.x + threadIdx.x;
  if (i < n4) dst[i] = src[i];
}

// ---------------------- edge scatter-sum (wave32 per edge) -----------------
// Each wave handles one edge: 32 lanes x float4 = 128 floats of nh[src],
// accumulated into sums[dst] with L2-resident f32 atomics.
__global__ void scatter_kernel(const float4* __restrict__ nh4,
                               const int* __restrict__ ei,
                               float* __restrict__ sums,
                               float* __restrict__ counts) {
  int t = blockIdx.x * blockDim.x + threadIdx.x;
  int e = t >> 5;
  int lane = t & 31;
  if (e >= N_EDGES) return;
  int src = ei[e];
  int dst = ei[N_EDGES + e];
  float4 v = nh4[(size_t)src * (D_IN / 4) + lane];
  float* p = sums + (size_t)dst * D_IN + lane * 4;
  atomicAdd(p + 0, v.x);
  atomicAdd(p + 1, v.y);
  atomicAdd(p + 2, v.z);
  atomicAdd(p + 3, v.w);
  if (lane == 0) atomicAdd(counts + dst, 1.0f);
}

// ---------------------- fused MLP on fp32 WMMA -----------------------------
// Block = 256 threads = 8 waves, handles 16 node rows.
// GEMM1: X[16x256] @ W1[256x256] -> each wave owns 2 N-tiles of 16 cols.
// GEMM2: H[16x256] @ W2[256x128] -> each wave owns 1 N-tile of 16 cols.
// A-frag (16x4 f32): lane<16 -> M=lane, K pair {k0,k0+1}; lane>=16 -> {k0+2,k0+3}.
// B-frag (4x16 f32): mirrored (N = lane%16, K half by lane>=16).
// C/D  (16x16 f32): VGPR j -> row j (lanes 0-15) / row j+8 (lanes 16-31), N = lane%16.
__global__ __launch_bounds__(256) void mlp_wmma_kernel(
    const float* __restrict__ nh, const float* __restrict__ sums,
    const float* __restrict__ counts,
    const float* __restrict__ W1, const float* __restrict__ b1,
    const float* __restrict__ W2, const float* __restrict__ b2,
    float* __restrict__ out) {
  __shared__ float X[16 * XPITCH];
  __shared__ float H[16 * XPITCH];
  __shared__ float invc[16];

  const int tid = threadIdx.x;
  const int base = blockIdx.x * 16;  // 50000 = 3125 * 16, no tail

  if (tid < 16) {
    float c = counts[base + tid];
    invc[tid] = 1.0f / fmaxf(c, 1.0f);
  }
  __syncthreads();

  // Stage concat input tile [16][256] = [nh | mean(agg)] into LDS.
  for (int idx = tid; idx < 16 * 256; idx += 256) {
    int row = idx >> 8;
    int col = idx & 255;
    int node = base + row;
    float v = (col < D_IN) ? nh[(size_t)node * D_IN + col]
                           : sums[(size_t)node * D_IN + (col - D_IN)] * invc[row];
    X[row * XPITCH + col] = v;
  }
  __syncthreads();

  const int wave = tid >> 5;
  const int lane = tid & 31;
  const int m = lane & 15;             // row within fragment
  const int kb = (lane >> 4) << 1;     // 0 or 2: K-half owned by this lane
  const int mbase = (lane >> 4) << 3;  // C/D row split: 0 or 8

  // ---- GEMM1: each wave covers columns [n0, n0+32) of W1 ----
  const int n0 = wave * 32;
  v8f c0 = {};
  v8f c1 = {};
#pragma unroll 4
  for (int k0 = 0; k0 < D_HID; k0 += 4) {
    v2f a;
    a.x = X[m * XPITCH + k0 + kb];
    a.y = X[m * XPITCH + k0 + kb + 1];
    const float* w1p = W1 + (size_t)(k0 + kb) * D_HID;
    v2f bA, bB;
    bA.x = w1p[n0 + m];
    bA.y = w1p[D_HID + n0 + m];
    bB.x = w1p[n0 + 16 + m];
    bB.y = w1p[D_HID + n0 + 16 + m];
    c0 = __builtin_amdgcn_wmma_f32_16x16x4_f32(false, a, false, bA, (short)0, c0, false, false);
    c1 = __builtin_amdgcn_wmma_f32_16x16x4_f32(false, a, false, bB, (short)0, c1, false, false);
  }
  {
    float bias0 = b1[n0 + m];
    float bias1 = b1[n0 + 16 + m];
#pragma unroll
    for (int j = 0; j < 8; ++j) {
      H[(mbase + j) * XPITCH + n0 + m]      = fmaxf(c0[j] + bias0, 0.0f);
      H[(mbase + j) * XPITCH + n0 + 16 + m] = fmaxf(c1[j] + bias1, 0.0f);
    }
  }
  __syncthreads();

  // ---- GEMM2: each wave covers columns [n2, n2+16) of W2 ----
  const int n2 = wave * 16;
  v8f c2 = {};
#pragma unroll 4
  for (int k0 = 0; k0 < D_HID; k0 += 4) {
    v2f a;
    a.x = H[m * XPITCH + k0 + kb];
    a.y = H[m * XPITCH + k0 + kb + 1];
    const float* w2p = W2 + (size_t)(k0 + kb) * D_OUT;
    v2f b;
    b.x = w2p[n2 + m];
    b.y = w2p[D_OUT + n2 + m];
    c2 = __builtin_amdgcn_wmma_f32_16x16x4_f32(false, a, false, b, (short)0, c2, false, false);
  }
  {
    float bias = b2[n2 + m];
#pragma unroll
    for (int j = 0; j < 8; ++j) {
      out[(size_t)(base + mbase + j) * D_OUT + n2 + m] = c2[j] + bias;
    }
  }
}

// ---------------------------------------------------------------------------
extern "C" void kernel_launch(void* const* d_in, const int* in_sizes, int n_in,
                              void* d_out, int out_size, void* d_ws, size_t ws_size,
                              hipStream_t stream) {
  const float* nh = (const float*)d_in[0];
  const float* eh = (const float*)d_in[1];
  const float* W1 = (const float*)d_in[2];
  const float* b1 = (const float*)d_in[3];
  const float* W2 = (const float*)d_in[4];
  const float* b2 = (const float*)d_in[5];
  const int* ei   = (const int*)d_in[6];
  float* out = (float*)d_out;

  float* sums   = (float*)d_ws;                      // N_NODES * D_IN
  float* counts = sums + (size_t)N_NODES * D_IN;     // N_NODES

  // 1) zero the scatter accumulators
  int nzero = N_NODES * D_IN + N_NODES;
  hipLaunchKernelGGL(zero_ws_kernel, dim3((nzero + 255) / 256), dim3(256), 0, stream,
                     sums, nzero);

  // 2) stream eh -> second output region (independent of scatter)
  int n4 = (N_EDGES * D_IN) / 4;
  hipLaunchKernelGGL(copy_eh_kernel, dim3((n4 + 255) / 256), dim3(256), 0, stream,
                     (const float4*)eh, (float4*)(out + (size_t)N_NODES * D_OUT), n4);

  // 3) edge scatter-sum: one wave32 per edge
  hipLaunchKernelGGL(scatter_kernel, dim3((N_EDGES * 32) / 256), dim3(256), 0, stream,
                     (const float4*)nh, ei, sums, counts);

  // 4) fused mean + concat + 2-layer MLP on fp32 WMMA
  hipLaunchKernelGGL(mlp_wmma_kernel, dim3(N_NODES / 16), dim3(256), 0, stream,
                     nh, sums, counts, W1, b1, W2, b2, out);
}